// DescriptorLayer_87917980549430
// MI455X (gfx1250) — compile-verified
//
#include <hip/hip_runtime.h>
#include <math.h>

#define H 512
#define W 512
#define BATCH 32
#define PS (H*W)
#define NPIX (BATCH*PS)
#define NB 4096

typedef __attribute__((ext_vector_type(2))) float v2f;
typedef __attribute__((ext_vector_type(8))) float v8f;

enum { PAD_ZERO = 0, PAD_SYMM = 1, PAD_R101 = 2 };

__device__ __forceinline__ v8f wmma4(v2f a, v2f b, v8f c) {
  // D(16x16,f32) = A(16x4,f32) * B(4x16,f32) + C  -> v_wmma_f32_16x16x4_f32
  return __builtin_amdgcn_wmma_f32_16x16x4_f32(
      /*neg_a=*/false, a, /*neg_b=*/false, b,
      /*c_mod=*/(short)0, c, /*reuse_a=*/false, /*reuse_b=*/false);
}

__device__ __forceinline__ int imin(int a, int b) { return a < b ? a : b; }
__device__ __forceinline__ int imax(int a, int b) { return a > b ? a : b; }

__device__ __forceinline__ int refl(int i, int n, int mode) {
  if (mode == PAD_SYMM)      { if (i < 0) i = -i - 1; else if (i >= n) i = 2*n - 1 - i; }
  else if (mode == PAD_R101) { if (i < 0) i = -i;     else if (i >= n) i = 2*n - 2 - i; }
  return i;
}

// Branchless padded load: always issues the global load (clamped address),
// selects 0 for out-of-range in zero-pad mode. No exec-mask divergence.
template<int PADM>
__device__ __forceinline__ float load_pad(const float* __restrict__ p, int i, int j) {
  if (PADM == PAD_ZERO) {
    bool ok = ((unsigned)i < (unsigned)H) & ((unsigned)j < (unsigned)W);
    int ii = imin(imax(i, 0), H - 1);
    int jj = imin(imax(j, 0), W - 1);
    float v = p[ii * W + jj];
    return ok ? v : 0.f;
  } else {
    int ii = refl(i, H, PADM);
    int jj = refl(j, W, PADM);
    return p[ii * W + jj];
  }
}

// -------------------------------------------------------------------------
// Separable 1-D correlation pass via WMMA f32 16x16x4.
//   out(i,j) = sum_t taps[t] * in(i, j + t - R)       (horizontal, VERT=0)
//   out(i,j) = sum_t taps[t] * in(i + t - R, j)       (vertical,   VERT=1)
// Banded coefficient matrix K(s,n) = taps[s-n], held in a zero-guarded LDS
// array skg[64] (taps live at offset 16, zeros elsewhere) so every band
// fetch is one unconditional ds_load_b32. All operands for all KC chunks
// are loaded up front into register arrays, then the WMMAs run as a single
// back-to-back accumulator chain.
// One wave32 computes one 16x16 tile; 8 waves (256 threads) per block.
// TAPK: 0 = gaussian(sigma), 1 = box; taps normalized to sum 1.
// -------------------------------------------------------------------------
template<int L, int PADM, bool VERT, int TAPK>
__global__ void wmma_sep_conv(const float* __restrict__ src, float* __restrict__ dst,
                              float sigma) {
  constexpr int R    = L / 2;
  constexpr int SPAN = 16 + 2 * R;
  constexpr int KC   = (SPAN + 3) >> 2;

  __shared__ float skg[64];                 // zero-guarded band taps
  int tid = threadIdx.x;
  if (tid < 64) skg[tid] = 0.f;
  __syncthreads();
  if (tid < L) {
    float v = 1.0f;
    if (TAPK == 0) { float x = (float)(tid - R); v = expf(-0.5f * x * x / (sigma * sigma)); }
    skg[16 + tid] = v;
  }
  __syncthreads();
  if (tid == 0) {
    float s = 0.f;
    for (int t = 0; t < L; ++t) s += skg[16 + t];
    float inv = 1.f / s;
    for (int t = 0; t < L; ++t) skg[16 + t] *= inv;
  }
  __syncthreads();

  int wave = tid >> 5, lane = tid & 31;
  int half = lane >> 4, lidx = lane & 15;
  int tile = blockIdx.x * 8 + wave;        // 1024 tiles / image
  int ti = (tile >> 5) * 16;
  int tj = (tile & 31) * 16;
  const float* sp = src + (size_t)blockIdx.y * PS;
  float*       dp = dst + (size_t)blockIdx.y * PS;

  // ---- phase 1: fetch all operands (loads get clause-grouped) ----
  v2f A[KC], B[KC];
#pragma unroll
  for (int c = 0; c < KC; ++c) {
    int k0 = 4 * c + 2 * half;             // K-slot owned by this half-wave
    int t  = k0 - lidx;                    // band tap index (may be out of band -> 0)
    if (!VERT) {
      // A = image rows (M=lidx, K=k0..k0+1), B = band (K, N=lidx)
      A[c].x = load_pad<PADM>(sp, ti + lidx, tj - R + k0);
      A[c].y = load_pad<PADM>(sp, ti + lidx, tj - R + k0 + 1);
      B[c].x = skg[16 + t];
      B[c].y = skg[17 + t];
    } else {
      // A = band (M=lidx, K), B = image columns (K = input row, N=lidx)
      A[c].x = skg[16 + t];
      A[c].y = skg[17 + t];
      B[c].x = load_pad<PADM>(sp, ti - R + k0,     tj + lidx);
      B[c].y = load_pad<PADM>(sp, ti - R + k0 + 1, tj + lidx);
    }
  }

  // ---- phase 2: back-to-back WMMA accumulator chain ----
  v8f acc = {0.f, 0.f, 0.f, 0.f, 0.f, 0.f, 0.f, 0.f};
#pragma unroll
  for (int c = 0; c < KC; ++c) acc = wmma4(A[c], B[c], acc);

  // D layout: VGPR v: lanes 0-15 -> M=v, lanes 16-31 -> M=v+8; N = lidx
#pragma unroll
  for (int v = 0; v < 8; ++v) {
    int m = v + 8 * half;
    dp[(ti + m) * W + tj + lidx] = acc[v];
  }
}

// -------------------------------------------------------------------------
// Pointwise / stencil kernels
// -------------------------------------------------------------------------
__device__ __forceinline__ void unrav(int idx, int& img, int& i, int& j) {
  img = idx / PS; int r = idx - img * PS; i = r / W; j = r - i * W;
}

// np.gradient along an axis: central diff interior, one-sided edges
__global__ void k_grad(const float* __restrict__ s, float* __restrict__ g,
                       int axis, int n) {
  int idx = blockIdx.x * blockDim.x + threadIdx.x;
  if (idx >= n) return;
  int img, i, j; unrav(idx, img, i, j);
  const float* p = s + (size_t)img * PS;
  float v;
  if (axis == 0) {
    if (i == 0)          v = p[W + j] - p[j];
    else if (i == H - 1) v = p[(H-1)*W + j] - p[(H-2)*W + j];
    else                 v = 0.5f * (p[(i+1)*W + j] - p[(i-1)*W + j]);
  } else {
    if (j == 0)          v = p[i*W + 1] - p[i*W];
    else if (j == W - 1) v = p[i*W + W-1] - p[i*W + W-2];
    else                 v = 0.5f * (p[i*W + j+1] - p[i*W + j-1]);
  }
  g[idx] = v;
}

__device__ __forceinline__ float dax0(const float* p, int i, int j) {
  if (i == 0)     return p[W + j] - p[j];
  if (i == H - 1) return p[(H-1)*W + j] - p[(H-2)*W + j];
  return 0.5f * (p[(i+1)*W + j] - p[(i-1)*W + j]);
}
__device__ __forceinline__ float dax1(const float* p, int i, int j) {
  if (j == 0)     return p[i*W + 1] - p[i*W];
  if (j == W - 1) return p[i*W + W-1] - p[i*W + W-2];
  return 0.5f * (p[i*W + j+1] - p[i*W + j-1]);
}

// largest hessian eigenvalue from gr, gc planes -> out channel 0
__global__ void k_hessian(const float* __restrict__ gr, const float* __restrict__ gc,
                          float* __restrict__ out, int n) {
  int idx = blockIdx.x * blockDim.x + threadIdx.x;
  if (idx >= n) return;
  int img, i, j; unrav(idx, img, i, j);
  const float* pr = gr + (size_t)img * PS;
  const float* pc = gc + (size_t)img * PS;
  float Hrr = dax0(pr, i, j);
  float Hrc = dax1(pr, i, j);
  float Hcc = dax1(pc, i, j);
  float mid  = 0.5f * (Hrr + Hcc);
  float d    = 0.5f * (Hrr - Hcc);
  float disc = sqrtf(d * d + Hrc * Hrc);
  out[(size_t)idx * 5 + 0] = mid + disc;
}

__global__ void k_mul(const float* a, const float* b, float* o, int n) {
  int idx = blockIdx.x * blockDim.x + threadIdx.x;
  if (idx < n) o[idx] = a[idx] * b[idx];
}

__global__ void k_copy_ch(const float* p, float* out, float scale, int ch, int n) {
  int idx = blockIdx.x * blockDim.x + threadIdx.x;
  if (idx < n) out[(size_t)idx * 5 + ch] = p[idx] * scale;
}

__global__ void k_quantize(const float* x, float* y, int n) {
  int idx = blockIdx.x * blockDim.x + threadIdx.x;
  if (idx < n) y[idx] = fminf(fmaxf(floorf(x[idx] * 255.0f), 0.f), 255.f);
}

__global__ void k_round255(float* y, int n) {   // in-place: clip(rint(v),0,255)/255
  int idx = blockIdx.x * blockDim.x + threadIdx.x;
  if (idx < n) y[idx] = fminf(fmaxf(rintf(y[idx]), 0.f), 255.f) * (1.0f / 255.0f);
}

// masked-gaussian normalization: divide by gauss(ones) = nv(i)*nh(j) (zero pad)
template<int L>
__global__ void k_norm_div(float* __restrict__ sm, float sigma, int n) {
  constexpr int R = L / 2;
  __shared__ float sk[32];
  int tid = threadIdx.x;
  if (tid < L) { float x = (float)(tid - R); sk[tid] = expf(-0.5f * x * x / (sigma * sigma)); }
  __syncthreads();
  if (tid == 0) {
    float s = 0.f; for (int t = 0; t < L; ++t) s += sk[t];
    float inv = 1.f / s; for (int t = 0; t < L; ++t) sk[t] *= inv;
  }
  __syncthreads();
  int idx = blockIdx.x * blockDim.x + tid;
  if (idx >= n) return;
  int img, i, j; unrav(idx, img, i, j); (void)img;
  float nv = 0.f, nh = 0.f;
  for (int t = 0; t < L; ++t) {
    if ((unsigned)(i + t - R) < (unsigned)H) nv += sk[t];
    if ((unsigned)(j + t - R) < (unsigned)W) nh += sk[t];
  }
  sm[idx] = sm[idx] / (nv * nh + 1e-12f);
}

// 3x3 Sobel (correlation, symmetric pad): gi = d(x)s, gj = s(x)d, m = hypot
__global__ void k_sobel(const float* __restrict__ sm, float* __restrict__ gi,
                        float* __restrict__ gj, float* __restrict__ m, int n) {
  int idx = blockIdx.x * blockDim.x + threadIdx.x;
  if (idx >= n) return;
  int img, i, j; unrav(idx, img, i, j);
  const float* p = sm + (size_t)img * PS;
  const float d3[3] = {-1.f, 0.f, 1.f};
  const float s3[3] = { 1.f, 2.f, 1.f};
  float vi = 0.f, vj = 0.f;
#pragma unroll
  for (int a = 0; a < 3; ++a) {
    int ii = refl(i + a - 1, H, PAD_SYMM);
#pragma unroll
    for (int b = 0; b < 3; ++b) {
      int jj = refl(j + b - 1, W, PAD_SYMM);
      float v = p[ii * W + jj];
      vi += d3[a] * s3[b] * v;
      vj += s3[a] * d3[b] * v;
    }
  }
  gi[idx] = vi; gj[idx] = vj;
  m[idx] = sqrtf(vi * vi + vj * vj);
}

// -------------------------------------------------------------------------
// Percentile thresholds (histogram approximation of jnp.percentile)
// -------------------------------------------------------------------------
__global__ void k_zero(int* p, int n) {
  int idx = blockIdx.x * blockDim.x + threadIdx.x;
  if (idx < n) p[idx] = 0;
}

__global__ void k_max(const float* __restrict__ m, int* __restrict__ maxv) {
  __shared__ float red[256];
  int img = blockIdx.y;
  int idx = blockIdx.x * 256 + threadIdx.x;
  red[threadIdx.x] = m[(size_t)img * PS + idx];   // m >= 0
  __syncthreads();
  for (int s = 128; s > 0; s >>= 1) {
    if (threadIdx.x < s) red[threadIdx.x] = fmaxf(red[threadIdx.x], red[threadIdx.x + s]);
    __syncthreads();
  }
  if (threadIdx.x == 0) atomicMax(&maxv[img], __float_as_int(red[0])); // ok: non-negative floats
}

__global__ void k_hist(const float* __restrict__ m, const int* __restrict__ maxv,
                       int* __restrict__ hist) {
  __shared__ int hh[NB];
  for (int i = threadIdx.x; i < NB; i += 256) hh[i] = 0;
  __syncthreads();
  int img = blockIdx.y;
  int base = blockIdx.x * 4096;
  float mv = __int_as_float(maxv[img]);
  float sc = (mv > 0.f) ? (float)NB / mv : 0.f;
#pragma unroll
  for (int k = 0; k < 16; ++k) {
    float v = m[(size_t)img * PS + base + threadIdx.x + k * 256];
    int b = (int)(v * sc); b = b > NB - 1 ? NB - 1 : (b < 0 ? 0 : b);
    atomicAdd(&hh[b], 1);
  }
  __syncthreads();
  for (int i = threadIdx.x; i < NB; i += 256)
    if (hh[i]) atomicAdd(&hist[(size_t)img * NB + i], hh[i]);
}

__global__ void k_thresh(const int* __restrict__ hist, const int* __restrict__ maxv,
                         float* __restrict__ lowt, float* __restrict__ hight) {
  int img = blockIdx.x;
  if (threadIdx.x != 0) return;
  float mv = __int_as_float(maxv[img]);
  float tl = 0.20f * (float)(PS - 1);
  float th = 0.50f * (float)(PS - 1);
  long cum = 0; float lo = 0.f, hi = 0.f; int gl = 0;
  for (int b = 0; b < NB; ++b) {
    cum += hist[(size_t)img * NB + b];
    float v = mv * ((float)b + 0.5f) / (float)NB;
    if (!gl && (float)cum > tl) { lo = v; gl = 1; }
    if ((float)cum > th)        { hi = v; break; }
  }
  lowt[img] = lo; hight[img] = hi;
}

// -------------------------------------------------------------------------
// Interpolated non-maximum suppression + double threshold
// writes lowmask (0/1) and edges seed (= high mask)
// -------------------------------------------------------------------------
__global__ void k_nms(const float* __restrict__ gi, const float* __restrict__ gj,
                      const float* __restrict__ m, const float* __restrict__ lowt,
                      const float* __restrict__ hight, float* __restrict__ lowmask,
                      float* __restrict__ edges, int n) {
  int idx = blockIdx.x * blockDim.x + threadIdx.x;
  if (idx >= n) return;
  int img, i, j; unrav(idx, img, i, j);
  const float* mp = m + (size_t)img * PS;
  const float eps = 1e-12f;
  float vi = gi[idx], vj = gj[idx], mm = mp[i * W + j];
  float ai = fabsf(vi), aj = fabsf(vj);
  int si = vi >= 0.f ? 1 : -1;
  int sj = vj >= 0.f ? 1 : -1;
  auto at = [&](int di, int dj) -> float {
    int ii = i + di, jj = j + dj;
    if ((unsigned)ii >= (unsigned)H || (unsigned)jj >= (unsigned)W) return 0.f;
    return mp[ii * W + jj];
  };
  float wA = ai / fmaxf(aj, eps);
  float wB = aj / fmaxf(ai, eps);
  float npA = (1.f - wA) * at(0,  sj) + wA * at( si,  sj);
  float nmA = (1.f - wA) * at(0, -sj) + wA * at(-si, -sj);
  float npB = (1.f - wB) * at( si, 0) + wB * at( si,  sj);
  float nmB = (1.f - wB) * at(-si, 0) + wB * at(-si, -sj);
  bool isA = aj >= ai;
  float n_p = isA ? npA : npB;
  float n_m = isA ? nmA : nmB;
  bool interior = (i > 0) && (i < H - 1) && (j > 0) && (j < W - 1);
  bool lm   = interior && (mm > 0.f) && (mm >= n_p) && (mm >= n_m);
  bool low  = lm && (mm > lowt[img]);
  bool high = low && (mm > hight[img]);
  lowmask[idx] = low  ? 1.f : 0.f;
  edges[idx]   = high ? 1.f : 0.f;
}

// -------------------------------------------------------------------------
// Hysteresis: block-synchronous flood fill. Each block converges its 32x32
// tile (+1 halo) in LDS; outer iterations (host loop, ping-pong planes)
// propagate across tiles. Fixed outer count -> deterministic under capture.
// -------------------------------------------------------------------------
__global__ void k_hyst(const float* __restrict__ esrc, const float* __restrict__ low,
                       float* __restrict__ edst) {
  __shared__ float se[2][34 * 34];
  __shared__ float sl[34 * 34];
  int img = blockIdx.z;
  int ti = blockIdx.y * 32, tj = blockIdx.x * 32;
  const float* ep = esrc + (size_t)img * PS;
  const float* lp = low  + (size_t)img * PS;
  float*       dp = edst + (size_t)img * PS;
  int tid = threadIdx.x;
  for (int idx = tid; idx < 34 * 34; idx += 256) {
    int li = idx / 34, lj = idx - li * 34;
    int gi_ = ti + li - 1, gj_ = tj + lj - 1;
    float e = 0.f, l = 0.f;
    if ((unsigned)gi_ < (unsigned)H && (unsigned)gj_ < (unsigned)W) {
      e = ep[gi_ * W + gj_]; l = lp[gi_ * W + gj_];
    }
    se[0][idx] = e; se[1][idx] = e; sl[idx] = l;
  }
  __syncthreads();
  int cur = 0;
  for (int it = 0; it < 40; ++it) {
    int changed = 0;
    for (int idx = tid; idx < 1024; idx += 256) {
      int li = idx >> 5, lj = idx & 31;
      int p = (li + 1) * 34 + (lj + 1);
      float e = se[cur][p];
      float ne = e;
      if (sl[p] > 0.5f && e < 0.5f) {
        float s = se[cur][p-35] + se[cur][p-34] + se[cur][p-33] + se[cur][p-1] +
                  se[cur][p+1]  + se[cur][p+33] + se[cur][p+34] + se[cur][p+35];
        if (s > 0.f) { ne = 1.f; changed = 1; }
      }
      se[cur ^ 1][p] = ne;
    }
    int any = __syncthreads_count(changed);
    cur ^= 1;
    if (any == 0) break;
  }
  for (int idx = tid; idx < 1024; idx += 256) {
    int li = idx >> 5, lj = idx & 31;
    dp[(ti + li) * W + (tj + lj)] = se[cur][(li + 1) * 34 + (lj + 1)];
  }
}

// -------------------------------------------------------------------------
// Host-side orchestration
// -------------------------------------------------------------------------
extern "C" void kernel_launch(void* const* d_in, const int* in_sizes, int n_in,
                              void* d_out, int out_size, void* d_ws, size_t ws_size,
                              hipStream_t stream) {
  const float* x = (const float*)d_in[0];       // [32,512,512,1] == 32 planes
  float* out = (float*)d_out;                   // [32,512,512,5]
  char* ws = (char*)d_ws;

  // workspace header (re-initialized every call -> deterministic)
  int*   maxv  = (int*)ws;                      // 32 ints (float bits, init 0)
  float* lowt  = (float*)(ws + 128);
  float* hight = (float*)(ws + 256);
  int*   hist  = (int*)(ws + 512);              // 32 * 4096 ints
  float* T0 = (float*)(ws + (1 << 20));
  float* T1 = T0 + (size_t)BATCH * PS;
  float* T2 = T1 + (size_t)BATCH * PS;
  float* T3 = T2 + (size_t)BATCH * PS;

  dim3 cblk(256), cgrd(128, BATCH);             // 8 wave-tiles per block
  dim3 pblk(256), pgrd((NPIX + 255) / 256);

  // ---- 1. hessian eigenvalue (out ch 0) ----
  wmma_sep_conv<19, PAD_ZERO, false, 0><<<cgrd, cblk, 0, stream>>>(x,  T0, 2.2f);
  wmma_sep_conv<19, PAD_ZERO, true,  0><<<cgrd, cblk, 0, stream>>>(T0, T1, 2.2f);
  k_grad<<<pgrd, pblk, 0, stream>>>(T1, T2, 0, NPIX);
  k_grad<<<pgrd, pblk, 0, stream>>>(T1, T3, 1, NPIX);
  k_hessian<<<pgrd, pblk, 0, stream>>>(T2, T3, out, NPIX);

  // ---- 2. structure tensor (out ch 1..3) ----
  k_grad<<<pgrd, pblk, 0, stream>>>(x, T0, 0, NPIX);   // gr
  k_grad<<<pgrd, pblk, 0, stream>>>(x, T1, 1, NPIX);   // gc
  k_mul<<<pgrd, pblk, 0, stream>>>(T0, T0, T2, NPIX);
  wmma_sep_conv<25, PAD_SYMM, false, 0><<<cgrd, cblk, 0, stream>>>(T2, T3, 3.0f);
  wmma_sep_conv<25, PAD_SYMM, true,  0><<<cgrd, cblk, 0, stream>>>(T3, T2, 3.0f);
  k_copy_ch<<<pgrd, pblk, 0, stream>>>(T2, out, 0.25f, 1, NPIX);
  k_mul<<<pgrd, pblk, 0, stream>>>(T0, T1, T2, NPIX);
  wmma_sep_conv<25, PAD_SYMM, false, 0><<<cgrd, cblk, 0, stream>>>(T2, T3, 3.0f);
  wmma_sep_conv<25, PAD_SYMM, true,  0><<<cgrd, cblk, 0, stream>>>(T3, T2, 3.0f);
  k_copy_ch<<<pgrd, pblk, 0, stream>>>(T2, out, 0.25f, 2, NPIX);
  k_mul<<<pgrd, pblk, 0, stream>>>(T1, T1, T2, NPIX);
  wmma_sep_conv<25, PAD_SYMM, false, 0><<<cgrd, cblk, 0, stream>>>(T2, T3, 3.0f);
  wmma_sep_conv<25, PAD_SYMM, true,  0><<<cgrd, cblk, 0, stream>>>(T3, T2, 3.0f);
  k_copy_ch<<<pgrd, pblk, 0, stream>>>(T2, out, 0.25f, 3, NPIX);

  // ---- 3. canny (out ch 4) ----
  k_quantize<<<pgrd, pblk, 0, stream>>>(x, T0, NPIX);
  wmma_sep_conv<5, PAD_R101, false, 1><<<cgrd, cblk, 0, stream>>>(T0, T1, 0.f);
  wmma_sep_conv<5, PAD_R101, true,  1><<<cgrd, cblk, 0, stream>>>(T1, T2, 0.f);
  k_round255<<<pgrd, pblk, 0, stream>>>(T2, NPIX);
  wmma_sep_conv<17, PAD_ZERO, false, 0><<<cgrd, cblk, 0, stream>>>(T2, T0, 2.0f);
  wmma_sep_conv<17, PAD_ZERO, true,  0><<<cgrd, cblk, 0, stream>>>(T0, T1, 2.0f);
  k_norm_div<17><<<pgrd, pblk, 0, stream>>>(T1, 2.0f, NPIX);
  k_sobel<<<pgrd, pblk, 0, stream>>>(T1, T0, T2, T3, NPIX);   // gi=T0 gj=T2 m=T3

  int hdr_ints = 128 + BATCH * NB;       // maxv..hight region (as ints) + hist
  k_zero<<<(hdr_ints + 255) / 256, 256, 0, stream>>>((int*)ws, hdr_ints);
  k_max<<<dim3(PS / 256, BATCH), 256, 0, stream>>>(T3, maxv);
  k_hist<<<dim3(PS / 4096, BATCH), 256, 0, stream>>>(T3, maxv, hist);
  k_thresh<<<dim3(BATCH), 32, 0, stream>>>(hist, maxv, lowt, hight);

  k_nms<<<pgrd, pblk, 0, stream>>>(T0, T2, T3, lowt, hight, T1, T0, NPIX);

  dim3 hgrd(16, 16, BATCH);
  for (int it = 0; it < 20; ++it) {      // even count: result ends in T0
    const float* s = (it & 1) ? T2 : T0;
    float*       d = (it & 1) ? T0 : T2;
    k_hyst<<<hgrd, 256, 0, stream>>>(s, T1, d);
  }
  k_copy_ch<<<pgrd, pblk, 0, stream>>>(T0, out, 1.0f, 4, NPIX);

  (void)in_sizes; (void)n_in; (void)out_size; (void)ws_size;
}